// IterateRightLayer_20289425506633
// MI455X (gfx1250) — compile-verified
//
#include <hip/hip_runtime.h>

#define CODE_LEN 1024
#define NUM_STAGES 10
#define NUM_LAYERS (NUM_STAGES + 1)
#define HALF 512
#define CLIP_V 20.0f

typedef float v2f __attribute__((ext_vector_type(2)));

__device__ __forceinline__ float min_sum(float x, float y) {
    // sign(x)*sign(y)*min(|x|,|y|): magnitude-min with XOR'd sign bit.
    float m = fminf(fabsf(x), fabsf(y));
    unsigned s = (__float_as_uint(x) ^ __float_as_uint(y)) & 0x80000000u;
    return __uint_as_float(__float_as_uint(m) | s);
}

__device__ __forceinline__ float clipf(float v) {
    return fminf(fmaxf(v, -CLIP_V), CLIP_V);   // lowers to v_med3_num_f32
}

__global__ __launch_bounds__(HALF) void polar_right_update_kernel(
    const float* __restrict__ right,
    const float* __restrict__ left,
    const float* __restrict__ rw,
    const int* __restrict__ iter_p,
    float* __restrict__ out)
{
    __shared__ float cur[CODE_LEN];      // current right layer (unclipped)
    __shared__ float lbuf[2][CODE_LEN];  // ping-pong staged left layers

    const int b = blockIdx.x;
    const int t = threadIdx.x;           // 0..511, one butterfly pair per thread
    const long row = (long)b * NUM_LAYERS * CODE_LEN;
    const float* rightRow = right + row;
    const float* leftRow  = left  + row;
    float*       outRow   = out   + row;

    const int    iter = *iter_p;
    const float* wb   = rw + iter * NUM_STAGES * 2;

    // ---- prologue: right layer 0 -> cur + clipped out; left layer 1 -> lbuf[0]
    {
        v2f r0 = __builtin_nontemporal_load((const v2f*)(rightRow + 2 * t));
        cur[2 * t]     = r0.x;
        cur[2 * t + 1] = r0.y;
        v2f o0 = { clipf(r0.x), clipf(r0.y) };
        __builtin_nontemporal_store(o0, (v2f*)(outRow + 2 * t));

        v2f l1 = __builtin_nontemporal_load(
            (const v2f*)(leftRow + 1 * CODE_LEN + 2 * t));
        lbuf[0][2 * t]     = l1.x;
        lbuf[0][2 * t + 1] = l1.y;
    }
    __syncthreads();

    #pragma unroll
    for (int i = 0; i < NUM_STAGES; ++i) {
        const int pp = i & 1;            // ping-pong selector

        // Early-issue the next left layer (registers) to hide HBM latency.
        v2f lnext = { 0.f, 0.f };
        if (i + 2 <= NUM_STAGES) {
            lnext = __builtin_nontemporal_load(
                (const v2f*)(leftRow + (i + 2) * CODE_LEN + 2 * t));
        }
        // gfx1250 prefetch (global_prefetch_b8) for the layer after that.
        if (i + 3 <= NUM_STAGES) {
            __builtin_prefetch(leftRow + (i + 3) * CODE_LEN + 2 * t, 0, 1);
        }

        const int stride = 1 << i;
        const int j0 = ((t >> i) << (i + 1)) | (t & (stride - 1)); // bit i == 0 (POS)
        const int j1 = j0 + stride;                                 // bit i == 1 (NEG)

        const float rp1 = cur[j0];        // right_prev1 (m)
        const float rp0 = cur[j1];        // right_prev0 (nm)
        const float lp1 = lbuf[pp][j0];   // left_prev1  (m)
        const float lp0 = lbuf[pp][j1];   // left_prev0  (nm)

        const float w0 = wb[2 * i];
        const float w1 = wb[2 * i + 1];

        const float n0 = w0 * min_sum(rp1, lp0 + rp0);   // new_m  -> j0
        const float n1 = w1 * min_sum(rp1, lp1) + rp0;   // new_nm -> j1

        // Exclusive pair ownership: each LDS slot read+written by one thread.
        cur[j0] = n0;
        cur[j1] = n1;

        // Install next left layer into the other ping-pong buffer (unread this stage).
        if (i + 2 <= NUM_STAGES) {
            lbuf[pp ^ 1][2 * t]     = lnext.x;
            lbuf[pp ^ 1][2 * t + 1] = lnext.y;
        }

        // Store clipped layer i+1 directly from this thread's results.
        float* outL = outRow + (i + 1) * CODE_LEN;
        __builtin_nontemporal_store(clipf(n0), outL + j0);
        __builtin_nontemporal_store(clipf(n1), outL + j1);

        __syncthreads();   // single barrier per stage
    }
}

extern "C" void kernel_launch(void* const* d_in, const int* in_sizes, int n_in,
                              void* d_out, int out_size, void* d_ws, size_t ws_size,
                              hipStream_t stream) {
    const float* right = (const float*)d_in[0];
    const float* left  = (const float*)d_in[1];
    const float* rw    = (const float*)d_in[2];
    const int*   iter  = (const int*)d_in[3];
    float*       out   = (float*)d_out;

    const int batch = in_sizes[0] / (NUM_LAYERS * CODE_LEN);

    polar_right_update_kernel<<<batch, HALF, 0, stream>>>(right, left, rw, iter, out);
}